// GNNConv_37014028156990
// MI455X (gfx1250) — compile-verified
//
#include <hip/hip_runtime.h>
#include <hip/hip_bf16.h>

#define N_NODES 100000
#define E_EDGES 1600000
#define DIM     128

typedef __attribute__((ext_vector_type(2))) float v2f;
typedef __attribute__((ext_vector_type(8))) float v8f;

// ---------------------------------------------------------------------------
// Kernel 1: zero the scatter accumulators (agg[N,128] and cnt[N])
// ---------------------------------------------------------------------------
__global__ __launch_bounds__(256)
void zero_ws_kernel(float4* __restrict__ agg4, float* __restrict__ cnt) {
    const size_t stride = (size_t)gridDim.x * blockDim.x;
    size_t i = (size_t)blockIdx.x * blockDim.x + threadIdx.x;
    const size_t n4 = (size_t)N_NODES * (DIM / 4);
    for (size_t j = i; j < n4; j += stride) {
        agg4[j] = make_float4(0.f, 0.f, 0.f, 0.f);
    }
    for (size_t j = i; j < (size_t)N_NODES; j += stride) {
        cnt[j] = 0.f;
    }
}

// ---------------------------------------------------------------------------
// Kernel 2: h = relu(layernorm(x))  — one wave32 per row, float4 per lane
// ---------------------------------------------------------------------------
__global__ __launch_bounds__(256)
void ln_relu_kernel(const float* __restrict__ x,
                    const float* __restrict__ lnw,
                    const float* __restrict__ lnb,
                    float* __restrict__ h) {
    const int lane = threadIdx.x & 31;
    const int wave = threadIdx.x >> 5;
    const int row  = blockIdx.x * 8 + wave;
    if (row >= N_NODES) return;

    float4 v = reinterpret_cast<const float4*>(x)[(size_t)row * (DIM / 4) + lane];

    // mean over 128 elems (wave32 reduction: width 32 on gfx1250)
    float s = v.x + v.y + v.z + v.w;
    for (int off = 16; off > 0; off >>= 1) s += __shfl_xor(s, off, 32);
    const float mu = s * (1.0f / 128.0f);

    const float dx = v.x - mu, dy = v.y - mu, dz = v.z - mu, dw = v.w - mu;
    float q = dx * dx + dy * dy + dz * dz + dw * dw;
    for (int off = 16; off > 0; off >>= 1) q += __shfl_xor(q, off, 32);
    const float rs = rsqrtf(q * (1.0f / 128.0f) + 1e-5f);

    const int c = lane * 4;
    float4 o;
    o.x = fmaxf(dx * rs * lnw[c + 0] + lnb[c + 0], 0.f);
    o.y = fmaxf(dy * rs * lnw[c + 1] + lnb[c + 1], 0.f);
    o.z = fmaxf(dz * rs * lnw[c + 2] + lnb[c + 2], 0.f);
    o.w = fmaxf(dw * rs * lnw[c + 3] + lnb[c + 3], 0.f);
    reinterpret_cast<float4*>(h)[(size_t)row * (DIM / 4) + lane] = o;
}

// ---------------------------------------------------------------------------
// Kernel 3: edge scatter — agg[dst] += h[src], cnt[dst] += 1
// 128 threads per block = one feature lane per edge; grid-stride over edges.
// fp32 atomics resolve in L2 (agg/h are both L2-resident: 51 MB each vs 192 MB).
// ---------------------------------------------------------------------------
__global__ __launch_bounds__(128)
void scatter_mean_kernel(const float* __restrict__ h,
                         const long long* __restrict__ edge_index, // [2,E] i64
                         float* __restrict__ agg,
                         float* __restrict__ cnt,
                         int num_edges) {
    const int t = threadIdx.x; // feature 0..127
    for (int e = blockIdx.x; e < num_edges; e += gridDim.x) {
        const int s = (int)edge_index[e];                       // src (x_j)
        const int d = (int)edge_index[(size_t)num_edges + e];   // dst (target)
        const float v = h[(size_t)s * DIM + t];
        atomicAdd(agg + (size_t)d * DIM + t, v);
        if (t == 0) atomicAdd(cnt + d, 1.0f);
    }
}

// ---------------------------------------------------------------------------
// Kernel 4: out = (agg/max(cnt,1)) @ W_l^T + b_l + h @ W_r^T
//
// Block: 256 threads (8 waves) covering 16 rows x 128 cols; wave w owns the
// 16x16 tile at columns [16w, 16w+16).
//
// LDS layout: v2f sW[kq][half][n] (64 KB) where entry = W[n][4kq+2*half .. +1].
// Each lane ds_load_b64's exactly its WMMA B-pair (no VALU selects). The
// half=1 sub-block is rotated by 16 columns so lanes 0..15 and 16..31 hit
// disjoint LDS bank halves (conflict-free). The rotated column index is
// loop-invariant.
//
// Two accumulators: acc0 = h @ W_r^T, acc1 = agg @ W_l^T. The mean-divide
// 1/max(cnt,1) depends only on the output row, so it commutes with the K
// reduction and is applied in the epilogue — the inner loop is purely
// load + load + v_wmma_f32_16x16x4_f32.
// ---------------------------------------------------------------------------
__global__ __launch_bounds__(256)
void sage_gemm_kernel(const float* __restrict__ h,
                      const float* __restrict__ agg,
                      const float* __restrict__ cnt,
                      const float* __restrict__ W_l,   // [128,128] row-major
                      const float* __restrict__ W_r,   // [128,128] row-major
                      const float* __restrict__ b_l,   // [128]
                      float* __restrict__ out) {
    __shared__ v2f sW[2 * (DIM / 4) * DIM];  // [kq][half][n], 64 KB

    const int lane   = threadIdx.x & 31;
    const int wave   = threadIdx.x >> 5;
    const int half   = lane >> 4;        // 0 => K pair {0,1}; 1 => K pair {2,3}
    const int l16    = lane & 15;
    const int tile_m = blockIdx.x * 16;
    const int mrow   = tile_m + l16;     // this lane's A-matrix row
    const int ocol   = wave * 16 + l16;  // this lane's B/D column

    // loop-invariant, bank-separating B read index
    const int bidx = half * DIM + ((ocol + half * 16) & (DIM - 1));

    v8f acc0 = {};   // h   @ W_r^T
    v8f acc1 = {};   // agg @ W_l^T  (unscaled; mean-divide in epilogue)

    for (int phase = 0; phase < 2; ++phase) {
        const float* W = (phase == 0) ? W_r : W_l;
        const float* A = (phase == 0) ? h   : agg;

        if (phase) __syncthreads();                 // protect sW reuse
        // stage: split each float4 W[n][4kq..4kq+3] into two v2f halves,
        // rotating the hi half by 16 columns (bank separation on read)
        for (int i = threadIdx.x; i < (DIM / 4) * DIM; i += 256) {
            const int n  = i & (DIM - 1);
            const int kq = i >> 7;
            const float4 f =
                reinterpret_cast<const float4*>(W)[(size_t)n * (DIM / 4) + kq];
            v2f lo, hi;
            lo.x = f.x; lo.y = f.y;
            hi.x = f.z; hi.y = f.w;
            sW[(kq * 2 + 0) * DIM + n] = lo;
            sW[(kq * 2 + 1) * DIM + ((n + 16) & (DIM - 1))] = hi;
        }
        __syncthreads();

        const float* arow = A + (size_t)mrow * DIM + half * 2;
        v8f acc = (phase == 0) ? acc0 : acc1;
        #pragma unroll 4
        for (int kq = 0; kq < DIM / 4; ++kq) {
            const v2f b = sW[kq * 2 * DIM + bidx];
            v2f a;
            a.x = arow[kq * 4 + 0];
            a.y = arow[kq * 4 + 1];
            acc = __builtin_amdgcn_wmma_f32_16x16x4_f32(
                /*neg_a=*/false, a, /*neg_b=*/false, b,
                /*c_mod=*/(short)0, acc, /*reuse_a=*/false, /*reuse_b=*/false);
        }
        if (phase == 0) acc0 = acc; else acc1 = acc;
    }

    // D layout (16x16 f32): VGPR r -> lanes 0..15 M=r, lanes 16..31 M=8+r
    const float bias = b_l[ocol];
    #pragma unroll
    for (int r = 0; r < 8; ++r) {
        const int orow = tile_m + half * 8 + r;
        const float invr = 1.0f / fmaxf(cnt[orow], 1.0f);
        out[(size_t)orow * DIM + ocol] = acc0[r] + acc1[r] * invr + bias;
    }
}

// ---------------------------------------------------------------------------
// Launch: zero -> layernorm/relu -> edge scatter -> fused dual-GEMM
// Workspace layout: h[N*128] | agg[N*128] | cnt[N]  (~103 MB)
// ---------------------------------------------------------------------------
extern "C" void kernel_launch(void* const* d_in, const int* in_sizes, int n_in,
                              void* d_out, int out_size, void* d_ws, size_t ws_size,
                              hipStream_t stream) {
    const float*     x    = (const float*)d_in[0];
    const long long* ei   = (const long long*)d_in[1];   // int64 [2,E]
    const float*     ln_w = (const float*)d_in[2];
    const float*     ln_b = (const float*)d_in[3];
    const float*     W_l  = (const float*)d_in[4];
    const float*     b_l  = (const float*)d_in[5];
    const float*     W_r  = (const float*)d_in[6];
    float*           out  = (float*)d_out;

    const int num_edges = in_sizes[1] / 2;   // == E_EDGES

    float* h   = (float*)d_ws;
    float* agg = h + (size_t)N_NODES * DIM;
    float* cnt = agg + (size_t)N_NODES * DIM;

    zero_ws_kernel<<<2048, 256, 0, stream>>>((float4*)agg, cnt);
    ln_relu_kernel<<<N_NODES / 8, 256, 0, stream>>>(x, ln_w, ln_b, h);
    scatter_mean_kernel<<<16384, 128, 0, stream>>>(h, ei, agg, cnt, num_edges);
    sage_gemm_kernel<<<N_NODES / 16, 256, 0, stream>>>(h, agg, cnt, W_l, W_r, b_l, out);
}